// EdgeConv_54649163874410
// MI455X (gfx1250) — compile-verified
//
#include <hip/hip_runtime.h>

#define BATCH   32
#define NPTS    2048
#define DIM     16
#define K_NN    16
#define FILTERS 64

typedef float v2f __attribute__((ext_vector_type(2)));
typedef float v8f __attribute__((ext_vector_type(8)));

// Single-instruction cross-half exchange: ds_swizzle SWAPX16
// (group-of-32 encoding: offset15=0, xor_mask=0x10, or_mask=0, and_mask=0x1f)
__device__ __forceinline__ float swz_xor16(float v) {
    return __int_as_float(__builtin_amdgcn_ds_swizzle(__float_as_int(v), 0x401F));
}

// ---------------------------------------------------------------------------
// Kernel 1: brute-force 2-D KNN (top-16, sorted ascending) per point.
// One block = 256 threads = one 256-point segment of one batch.
// The batch's 2048 (x,y) points are staged in LDS (16 KB of 320 KB/WGP).
// ---------------------------------------------------------------------------
__global__ __launch_bounds__(256) void knn_kernel(const float* __restrict__ x,
                                                  int* __restrict__ knn) {
    __shared__ float2 pts[NPTS];
    const int bb  = blockIdx.x >> 3;   // batch
    const int seg = blockIdx.x & 7;    // 256-point segment
    const int tid = threadIdx.x;

    for (int j = tid; j < NPTS; j += 256) {
        const float* px = x + (size_t)(bb * NPTS + j) * DIM;
        pts[j] = make_float2(px[0], px[1]);
    }
    __syncthreads();

    const int i = seg * 256 + tid;
    const float2 p = pts[i];

    float bd[K_NN];
    int   bi[K_NN];
#pragma unroll
    for (int t = 0; t < K_NN; ++t) { bd[t] = 3.4e38f; bi[t] = 0; }

    for (int j = 0; j < NPTS; ++j) {
        float2 q = pts[j];
        float dx = p.x - q.x;
        float dy = p.y - q.y;
        float d  = fmaf(dx, dx, dy * dy);
        if (d < bd[K_NN - 1]) {
            bd[K_NN - 1] = d; bi[K_NN - 1] = j;
            // one bubble pass keeps the array sorted (it was sorted before)
#pragma unroll
            for (int t = K_NN - 1; t > 0; --t) {
                if (bd[t] < bd[t - 1]) {
                    float td = bd[t]; bd[t] = bd[t - 1]; bd[t - 1] = td;
                    int   ti = bi[t]; bi[t] = bi[t - 1]; bi[t - 1] = ti;
                }
            }
        }
    }

    // vectorized index writeback: 4 x global_store_b128
    int4* o = (int4*)(knn + (size_t)(bb * NPTS + i) * K_NN);
    o[0] = make_int4(bi[0],  bi[1],  bi[2],  bi[3]);
    o[1] = make_int4(bi[4],  bi[5],  bi[6],  bi[7]);
    o[2] = make_int4(bi[8],  bi[9],  bi[10], bi[11]);
    o[3] = make_int4(bi[12], bi[13], bi[14], bi[15]);
}

// ---------------------------------------------------------------------------
// Kernel 2: fused gather + edge-conv (WMMA f32 16x16x4 chained 4x over K=16)
// + max-pool + concat. One wave32 per point; 8 waves (8 points) per block.
//   A row m  = x[idx[m]] - centre            (16x16, f32)
//   B tile   = W[:, 16t:16t+16]              (16x16 of 64 filters)
//   A frag:  lane = (k>=2 ? 16+m : m), vgpr = k&1   (ISA 16x4 f32 layout)
//   C/D:     lane L holds column L&15, rows v + 8*(L>>4)
// The 17th row (centre . W) rides along as scalar FMAs on the same bfrag,
// co-executing with the WMMA pipe; halves combined with one ds_swizzle.
// After the SWAP16 combine both halves hold identical per-column results,
// so h=0 lanes store tiles 0/1 and h=1 lanes tiles 2/3: full-wave coalesced
// stores, EXEC stays all-1s for the whole kernel (no saveexec regions).
// ---------------------------------------------------------------------------
__global__ __launch_bounds__(256) void edgeconv_kernel(const float* __restrict__ x,
                                                       const float* __restrict__ W,
                                                       const float* __restrict__ bias,
                                                       const int* __restrict__ knn,
                                                       float* __restrict__ out) {
    __shared__ float Wl[DIM * FILTERS];  // W staged: [k][f], 4 KB
    const int tid = threadIdx.x;
    for (int i = tid; i < DIM * FILTERS; i += 256) Wl[i] = W[i];
    __syncthreads();

    const int wave = tid >> 5;
    const int lane = tid & 31;
    const int p    = blockIdx.x * 8 + wave;   // global point id
    const int b    = p >> 11;                 // / NPTS
    const int m    = lane & 15;               // row id / filter column
    const int h    = lane >> 4;               // half select

    const float* cb = x + (size_t)p * DIM;
    const int nb_idx = knn[(size_t)p * K_NN + m];
    const float* nb = x + (size_t)(b * NPTS + nb_idx) * DIM;

    // Per-chunk fragments: only compile-time indices into these arrays.
    v2f afrag[4];   // (neigh - centre) at dims (k0, k0+1)
    v2f cv[4];      // centre at dims (k0, k0+1)
#pragma unroll
    for (int c = 0; c < 4; ++c) {
        const int k0 = 4 * c + 2 * h;
        float2 nv = *(const float2*)(nb + k0);
        float2 cc = *(const float2*)(cb + k0);
        cv[c].x = cc.x;        cv[c].y = cc.y;
        afrag[c].x = nv.x - cc.x;
        afrag[c].y = nv.y - cc.y;
    }

    float r[4];                               // pooled result per filter tile
#pragma unroll
    for (int t = 0; t < 4; ++t) {             // 4 filter tiles of 16
        const int f = t * 16 + m;
        v8f acc = {};
        float dcp = 0.0f;                     // centre-row partial (8 dims/lane)
#pragma unroll
        for (int c = 0; c < 4; ++c) {         // chain K = 16 as 4 x (K=4)
            const int k0 = 4 * c + 2 * h;
            v2f bfrag;
            bfrag.x = Wl[(k0    ) * FILTERS + f];
            bfrag.y = Wl[(k0 + 1) * FILTERS + f];
            acc = __builtin_amdgcn_wmma_f32_16x16x4_f32(
                false, afrag[c], false, bfrag, (short)0, acc, false, false);
            dcp = fmaf(cv[c].x, bfrag.x, dcp);
            dcp = fmaf(cv[c].y, bfrag.y, dcp);
        }
        // max over the 8 rows held by this lane
        float mx = fmaxf(fmaxf(fmaxf(acc[0], acc[1]), fmaxf(acc[2], acc[3])),
                         fmaxf(fmaxf(acc[4], acc[5]), fmaxf(acc[6], acc[7])));
        // fold in the other half: rows 8..15 (max) and centre dims (sum)
        mx = fmaxf(mx, swz_xor16(mx));
        float dc = dcp + swz_xor16(dcp);
        r[t] = fmaxf(mx, dc);                 // identical across both halves
    }

    const size_t obase = (size_t)p * (FILTERS + DIM);

    // Full-wave stores: h=0 lanes cover filters [0,32), h=1 lanes [32,64).
    const int f0 = h * 32 + m;
    const int f1 = f0 + 16;
    const float v0 = h ? r[2] : r[0];
    const float v1 = h ? r[3] : r[1];
    out[obase + f0] = v0 + bias[f0];
    out[obase + f1] = v1 + bias[f1];

    // concat passthrough of x: both halves write the same value (benign).
    out[obase + FILTERS + m] = cb[m];
}

// ---------------------------------------------------------------------------
extern "C" void kernel_launch(void* const* d_in, const int* in_sizes, int n_in,
                              void* d_out, int out_size, void* d_ws, size_t ws_size,
                              hipStream_t stream) {
    const float* x    = (const float*)d_in[0];
    const float* W    = (const float*)d_in[1];
    const float* bias = (const float*)d_in[2];
    int*   knn = (int*)d_ws;           // BATCH*NPTS*K_NN ints = 4 MB scratch
    float* out = (float*)d_out;

    knn_kernel<<<BATCH * 8, 256, 0, stream>>>(x, knn);
    edgeconv_kernel<<<(BATCH * NPTS) / 8, 256, 0, stream>>>(x, W, bias, knn, out);
}